// QLSTMGen024_65481071404625
// MI455X (gfx1250) — compile-verified
//
#include <hip/hip_runtime.h>

// Problem dimensions (from reference)
#define SEQ   512
#define BATCH 1024
#define D_IN  32
#define D_H   96
#define NQ    128           // = D_IN + D_H
#define ROWS  16            // batch rows per block (= WMMA M)
#define NWAVE 6             // one wave per 16-column N tile (6*16 = 96)
#define NTHREADS (NWAVE*32)
#define NKSTEP (NQ/4)       // 32 k-steps of K=4 for V_WMMA_F32_16X16X4_F32
#define NTILE (D_H/16)      // 6

typedef __attribute__((ext_vector_type(2))) float v2f;  // WMMA f32 A/B fragment
typedef __attribute__((ext_vector_type(8))) float v8f;  // WMMA f32 C/D fragment

// LDS layout (floats)
#define WFRAG_FLOATS (4*NTILE*NKSTEP*32*2)   // 49152 (all 4 weight mats, fragment-order)
#define QF_FLOATS    (NKSTEP*32*2)           // 2048  (q in A-fragment order)
#define HX_FLOATS    (ROWS*D_H)              // 1536
#define CX_FLOATS    (ROWS*D_H)              // 1536
#define ATTN_FLOATS  (ROWS)                  // 16
#define LDS_FLOATS   (WFRAG_FLOATS+QF_FLOATS+HX_FLOATS+CX_FLOATS+ATTN_FLOATS)

// Fast activations: single v_rcp_f32 instead of the IEEE v_div_scale fixup chain.
__device__ __forceinline__ float fast_sigmoid(float x) {
  return __builtin_amdgcn_rcpf(1.0f + __expf(-x));
}
__device__ __forceinline__ float fast_tanh(float x) {
  return 1.0f - 2.0f * __builtin_amdgcn_rcpf(__expf(2.0f * x) + 1.0f);
}

__global__ __launch_bounds__(NTHREADS)
void qlstm_fused_persistent(const float* __restrict__ X,      // (SEQ, B, 32)
                            const float* __restrict__ theta,  // (128)
                            const float* __restrict__ Wf, const float* __restrict__ bf,
                            const float* __restrict__ Wi, const float* __restrict__ bi,
                            const float* __restrict__ Wu, const float* __restrict__ bu,
                            const float* __restrict__ Wo, const float* __restrict__ bo,
                            float* __restrict__ Y,            // (SEQ, B, 96)
                            float* __restrict__ HX,           // (B, 96)
                            float* __restrict__ CX)           // (B, 96)
{
  extern __shared__ float lds[];
  float* wfrag = lds;                    // weights in WMMA-B fragment order
  float* qf    = wfrag + WFRAG_FLOATS;   // q in WMMA-A fragment order
  float* hx    = qf + QF_FLOATS;         // carried hidden state [16][96]
  float* cx    = hx + HX_FLOATS;         // carried cell state   [16][96]
  float* attn  = cx + CX_FLOATS;         // per-row attention scalar [16]

  const int tid  = threadIdx.x;
  const int wave = tid >> 5;
  const int lane = tid & 31;
  const int row0 = blockIdx.x * ROWS;    // this block's batch stripe

  // ---- one-time: stage weights into WMMA-B fragment order in LDS ----
  // B matrix is K x N = W^T.  Fragment for k-step kk, columns [j*16, j*16+16):
  //   lanes 0-15:  (W[n][4kk+0], W[n][4kk+1]),  n = j*16+lane
  //   lanes 16-31: (W[n][4kk+2], W[n][4kk+3]),  n = j*16+lane-16
  const float* Wg[4] = {Wf, Wi, Wu, Wo};
  for (int tup = wave; tup < 4 * NTILE * NKSTEP; tup += NWAVE) {
    const int kk = tup & (NKSTEP - 1);
    const int j  = (tup >> 5) % NTILE;
    const int g  = tup / (NTILE * NKSTEP);
    const int n  = j * 16 + (lane & 15);
    const int k  = kk * 4 + (lane >> 4) * 2;
    const float2 w = *(const float2*)(Wg[g] + (size_t)n * NQ + k);
    v2f wv; wv.x = w.x; wv.y = w.y;
    *(v2f*)(wfrag + ((size_t)tup * 32 + lane) * 2) = wv;
  }
  for (int i = tid; i < ROWS * D_H; i += NTHREADS) { hx[i] = 0.0f; cx[i] = 0.0f; }

  // per-lane theta for the scan: lane covers k = 4*lane .. 4*lane+3
  const float4 th = *(const float4*)(theta + 4 * lane);

  // this wave's GEMM column tile
  const int   ncol = wave * 16 + (lane & 15);
  const float bfv = bf[ncol], biv = bi[ncol], buv = bu[ncol], bov = bo[ncol];
  const float* wbF = wfrag + (size_t)(0 * NTILE + wave) * NKSTEP * 64;
  const float* wbI = wfrag + (size_t)(1 * NTILE + wave) * NKSTEP * 64;
  const float* wbU = wfrag + (size_t)(2 * NTILE + wave) * NKSTEP * 64;
  const float* wbO = wfrag + (size_t)(3 * NTILE + wave) * NKSTEP * 64;
  const int rbase = (lane >> 4) * 8;     // C/D rows: lanes 0-15 -> M=e, 16-31 -> M=e+8

  // prefetch first timestep's X stripe (one 128B line per row)
  if (tid < ROWS) {
    __builtin_prefetch(X + ((size_t)row0 + tid) * D_IN, 0, 3);
  }

  __syncthreads();

  for (int t = 0; t < SEQ; ++t) {
    // ---- phase 1: q = cumprod(cos([x_t, hx] + theta)) -> A-fragment layout ----
    for (int r = wave; r < ROWS; r += NWAVE) {
      const int kbase = 4 * lane;
      float4 v;
      if (kbase < D_IN) {
        v = *(const float4*)(X + ((size_t)t * BATCH + row0 + r) * D_IN + kbase);
      } else {
        v = *(const float4*)(hx + r * D_H + (kbase - D_IN));
      }
      const float c0 = __cosf(v.x + th.x);
      const float c1 = __cosf(v.y + th.y);
      const float c2 = __cosf(v.z + th.z);
      const float c3 = __cosf(v.w + th.w);
      float incl = c0 * c1 * c2 * c3;         // per-lane segment product
      #pragma unroll
      for (int d = 1; d < 32; d <<= 1) {      // wave32 inclusive multiply-scan
        float up = __shfl_up(incl, d, 32);
        if (lane >= d) incl *= up;
      }
      float excl = __shfl_up(incl, 1, 32);
      if (lane == 0) excl = 1.0f;
      const float q0 = excl * c0, q1 = q0 * c1, q2 = q1 * c2, q3 = q2 * c3;
      // A-fragment: VGPR pair per k-step kk=lane; rows in lanes, K halves split
      v2f lo; lo.x = q0; lo.y = q1;
      v2f hi; hi.x = q2; hi.y = q3;
      *(v2f*)(qf + (lane * 32 + r     ) * 2) = lo;
      *(v2f*)(qf + (lane * 32 + r + 16) * 2) = hi;
    }
    __syncthreads();

    // ---- phase 2: 4 fused GEMMs on the matrix cores (fp32 WMMA) ----
    v8f accF, accI, accU, accO;
    #pragma unroll
    for (int e = 0; e < 8; ++e) { accF[e] = bfv; accI[e] = biv; accU[e] = buv; accO[e] = bov; }
    #pragma unroll 4
    for (int kk = 0; kk < NKSTEP; ++kk) {
      const v2f a   = *(const v2f*)(qf  + (kk * 32 + lane) * 2);
      const v2f bF_ = *(const v2f*)(wbF + kk * 64 + lane * 2);
      const v2f bI_ = *(const v2f*)(wbI + kk * 64 + lane * 2);
      const v2f bU_ = *(const v2f*)(wbU + kk * 64 + lane * 2);
      const v2f bO_ = *(const v2f*)(wbO + kk * 64 + lane * 2);
      accF = __builtin_amdgcn_wmma_f32_16x16x4_f32(false, a, false, bF_, (short)0, accF, false, false);
      accI = __builtin_amdgcn_wmma_f32_16x16x4_f32(false, a, false, bI_, (short)0, accI, false, false);
      accU = __builtin_amdgcn_wmma_f32_16x16x4_f32(false, a, false, bU_, (short)0, accU, false, false);
      accO = __builtin_amdgcn_wmma_f32_16x16x4_f32(false, a, false, bO_, (short)0, accO, false, false);
    }

    // ---- phase 3: gates, cell + hidden update (each (row,col) owned by 1 lane) ----
    float hreg[8];
    #pragma unroll
    for (int e = 0; e < 8; ++e) {
      const int r = rbase + e;
      const float fg = fast_sigmoid(accF[e]);
      const float ig = fast_sigmoid(accI[e]);
      const float gg = fast_tanh(accU[e]);
      const float og = fast_sigmoid(accO[e]);
      const float cnew = fg * cx[r * D_H + ncol] + ig * gg;
      cx[r * D_H + ncol] = cnew;
      const float h = og * fast_tanh(cnew);
      hreg[e] = h;
      hx[r * D_H + ncol] = h;     // pre-attention value (column 0 feeds attn)
    }
    __syncthreads();

    if (tid < ROWS) attn[tid] = __cosf(hx[tid * D_H]);
    __syncthreads();

    #pragma unroll
    for (int e = 0; e < 8; ++e) {
      const int r = rbase + e;
      hx[r * D_H + ncol] = hreg[e] * attn[r];   // post-attention carried state
    }
    __syncthreads();

    // ---- phase 4: prefetch next X stripe, then coalesced streaming store ----
    if (t + 1 < SEQ && tid < ROWS) {
      __builtin_prefetch(X + ((size_t)(t + 1) * BATCH + row0 + tid) * D_IN, 0, 3);
    }
    float* dst = Y + ((size_t)t * BATCH + row0) * D_H;
    for (int i = tid; i < ROWS * D_H; i += NTHREADS) {
      __builtin_nontemporal_store(hx[i], dst + i);   // write-once stream, TH=NT
    }
    // no barrier needed: next iteration only reads hx / writes qf, and hx is
    // next written after the phase-1 barrier of the next iteration.
  }

  __syncthreads();
  // final carried state
  for (int i = tid; i < ROWS * D_H; i += NTHREADS) {
    HX[(size_t)row0 * D_H + i] = hx[i];
    CX[(size_t)row0 * D_H + i] = cx[i];
  }
}

extern "C" void kernel_launch(void* const* d_in, const int* in_sizes, int n_in,
                              void* d_out, int out_size, void* d_ws, size_t ws_size,
                              hipStream_t stream) {
  (void)in_sizes; (void)n_in; (void)out_size; (void)d_ws; (void)ws_size;
  const float* X     = (const float*)d_in[0];
  const float* theta = (const float*)d_in[1];
  const float* Wf = (const float*)d_in[2]; const float* bf = (const float*)d_in[3];
  const float* Wi = (const float*)d_in[4]; const float* bi = (const float*)d_in[5];
  const float* Wu = (const float*)d_in[6]; const float* bu = (const float*)d_in[7];
  const float* Wo = (const float*)d_in[8]; const float* bo = (const float*)d_in[9];

  float* out = (float*)d_out;
  float* Y  = out;                                      // (SEQ, B, 96)
  float* HX = out + (size_t)SEQ * BATCH * D_H;          // (B, 96)
  float* CX = HX + (size_t)BATCH * D_H;                 // (B, 96)

  const size_t shmem = (size_t)LDS_FLOATS * sizeof(float);  // ~217 KB, < 320 KB/WG
  hipLaunchKernelGGL(qlstm_fused_persistent,
                     dim3(BATCH / ROWS), dim3(NTHREADS), shmem, stream,
                     X, theta, Wf, bf, Wi, bi, Wu, bu, Wo, bo, Y, HX, CX);
}